// BboxRegression_77824807403978
// MI455X (gfx1250) — compile-verified
//
#include <hip/hip_runtime.h>
#include <hip/hip_bf16.h>
#include <stdint.h>

typedef __attribute__((ext_vector_type(2))) float v2f;
typedef __attribute__((ext_vector_type(8))) float v8f;

#define BATCH 4096
#define NROWS 128
#define DDIM  512
#define ROWS_TOTAL (BATCH * NROWS)       // 524288
#define WAVES_PER_BLOCK 4
#define CHUNK_F 64                       // floats of D per staged chunk
#define NCHUNK (DDIM / CHUNK_F)          // 8
#define LDS_ROW 68                       // padded pitch (floats) -> conflict-free b64 reads
#define LDS_CHUNK (16 * LDS_ROW)         // 1088 floats per buffer

// ---------------------------------------------------------------------------
// Prep: expand W(512x4) into pre-swizzled padded pair table Wp.
// Pair index (kq in [0,128), half in {0,1}, m in [0,16)):
//   Wp[(kq*32 + half*16 + m)*2 + 0] = Wpad[4*kq + 2*half + 0][m]
//   Wp[(kq*32 + half*16 + m)*2 + 1] = Wpad[4*kq + 2*half + 1][m]
// where Wpad[k][m] = (m < 4) ? W[k*4+m] : 0.  Total 8192 floats = 32 KB.
// ---------------------------------------------------------------------------
__global__ void prep_wp_kernel(const float* __restrict__ W, float* __restrict__ Wp) {
  int t = blockIdx.x * blockDim.x + threadIdx.x;
  if (t >= 128 * 64) return;
  int lohi = t & 1;
  int pi   = t >> 1;
  int m    = pi & 15;
  int kqh  = pi >> 4;
  int half = kqh & 1;
  int kq   = kqh >> 1;
  int k    = kq * 4 + half * 2 + lohi;
  Wp[t] = (m < 4) ? W[k * 4 + m] : 0.0f;
}

// ---------------------------------------------------------------------------
// Argmax over ref_scores[b, :128]; also emits slice_inds (as floats).
// ---------------------------------------------------------------------------
__global__ void argmax_kernel(const float* __restrict__ scores,
                              int* __restrict__ inds,
                              float* __restrict__ out_slice) {
  int b = blockIdx.x * blockDim.x + threadIdx.x;
  if (b >= BATCH) return;
  const float* s = scores + (size_t)b * NROWS;
  float best = s[0];
  int bi = 0;
  for (int i = 1; i < NROWS; ++i) {
    float v = s[i];
    if (v > best) { best = v; bi = i; }   // strict > => first max (jnp.argmax)
  }
  inds[b] = bi;
  out_slice[(size_t)b * 2 + 0] = (float)b;
  out_slice[(size_t)b * 2 + 1] = (float)bi;
}

// ---------------------------------------------------------------------------
// Main GEMM: out_fcn(524288 x 4) = x(524288 x 512) @ W(512 x 4) + b
// One wave per 16-row tile. A staged via async global->LDS (double buffered),
// consumed by V_WMMA_F32_16X16X4_F32 (K stepped by 4, 128 WMMAs/tile).
// ---------------------------------------------------------------------------
__global__ __launch_bounds__(128) void gemm_wmma_kernel(
    const float* __restrict__ x, const float* __restrict__ Wp,
    const float* __restrict__ bias, float* __restrict__ out_fcn) {
  __shared__ float lds_buf[WAVES_PER_BLOCK * 2 * LDS_CHUNK];  // 34816 B

  const int wave    = threadIdx.x >> 5;
  const int lane    = threadIdx.x & 31;
  const int m       = lane & 15;      // A: row M / B,C: column N
  const int half    = lane >> 4;      // 0 or 1 (lane half)
  const int halfsel = half << 1;      // K sub-select: 0 or 2

  const size_t tile    = (size_t)blockIdx.x * WAVES_PER_BLOCK + wave;
  const size_t rowbase = tile * 16;
  const float* A = x + rowbase * DDIM;
  float* myLds = lds_buf + wave * 2 * LDS_CHUNK;

  // Issue one chunk: 16 async b64 copies, each moving one 256 B row-chunk
  // (32 lanes x 8 B, fully coalesced) into the padded LDS buffer.
  auto issue = [&](int c, int buf) {
    uint32_t lbase = (uint32_t)(uintptr_t)(myLds + buf * LDS_CHUNK);
    const float* g0 = A + c * CHUNK_F + lane * 2;
#pragma unroll
    for (int r = 0; r < 16; ++r) {
      uint32_t l  = lbase + (uint32_t)((r * LDS_ROW + lane * 2) * 4);
      uint64_t ga = (uint64_t)(uintptr_t)(g0 + r * DDIM);
      asm volatile("global_load_async_to_lds_b64 %0, %1, off"
                   :: "v"(l), "v"(ga)
                   : "memory");
    }
  };

  v8f acc = {0.f, 0.f, 0.f, 0.f, 0.f, 0.f, 0.f, 0.f};

  issue(0, 0);
  for (int c = 0; c < NCHUNK; ++c) {
    if (c + 1 < NCHUNK) {
      issue(c + 1, (c + 1) & 1);
      asm volatile("s_wait_asynccnt 16" ::: "memory");  // prior chunk complete
    } else {
      asm volatile("s_wait_asynccnt 0" ::: "memory");
    }
    const float* bufp = myLds + (c & 1) * LDS_CHUNK;
    const float* wp   = Wp + (size_t)c * 1024 + half * 32 + m * 2;
#pragma unroll
    for (int kk = 0; kk < 16; ++kk) {
      // A fragment (16x4 f32): lane<16 -> K=k,k+1 ; lane>=16 -> K=k+2,k+3
      v2f a  = *(const v2f*)(bufp + m * LDS_ROW + kk * 4 + halfsel);
      // B fragment (4x16 f32): pre-swizzled contiguous pair per lane
      v2f bv = *(const v2f*)(wp + kk * 64);
      acc = __builtin_amdgcn_wmma_f32_16x16x4_f32(
          /*neg_a=*/false, a, /*neg_b=*/false, bv,
          /*c_mod=*/(short)0, acc, /*reuse_a=*/false, /*reuse_b=*/false);
    }
  }

  // C/D layout: VGPR r -> M = r + 8*half, column N = m. Only N<4 is real.
  if (m < 4) {
    float bf = bias[m];
    const size_t row0 = rowbase + (size_t)half * 8;
#pragma unroll
    for (int r = 0; r < 8; ++r) {
      out_fcn[(row0 + r) * 4 + m] = acc[r] + bf;
    }
  }
}

// ---------------------------------------------------------------------------
// Gather bbox_offset[b, :] = bbox_offset_fcn[b, max_ind[b], :]
// ---------------------------------------------------------------------------
__global__ void gather_kernel(const float* __restrict__ fcn,
                              const int* __restrict__ inds,
                              float* __restrict__ out_off) {
  int t = blockIdx.x * blockDim.x + threadIdx.x;
  if (t >= BATCH * 4) return;
  int b = t >> 2;
  int f = t & 3;
  int ind = inds[b];
  out_off[t] = fcn[((size_t)b * NROWS + ind) * 4 + f];
}

extern "C" void kernel_launch(void* const* d_in, const int* in_sizes, int n_in,
                              void* d_out, int out_size, void* d_ws, size_t ws_size,
                              hipStream_t stream) {
  (void)in_sizes; (void)n_in; (void)out_size; (void)ws_size;
  const float* x      = (const float*)d_in[0];  // (B, N, D)
  const float* scores = (const float*)d_in[1];  // (B, N)
  const float* W      = (const float*)d_in[2];  // (D, 4)
  const float* bias   = (const float*)d_in[3];  // (4,)

  float* out       = (float*)d_out;
  float* out_off   = out;                                   // B*4     = 16384
  float* out_fcn   = out + (size_t)BATCH * 4;               // B*N*4   = 2097152
  float* out_slice = out + (size_t)BATCH * 4 + (size_t)BATCH * NROWS * 4;  // B*2

  int*   inds = (int*)d_ws;                                  // 16 KB
  float* Wp   = (float*)((char*)d_ws + 16384);               // 32 KB

  prep_wp_kernel<<<(128 * 64 + 255) / 256, 256, 0, stream>>>(W, Wp);
  argmax_kernel<<<(BATCH + 255) / 256, 256, 0, stream>>>(scores, inds, out_slice);

  const int tiles  = ROWS_TOTAL / 16;                        // 32768
  const int blocks = tiles / WAVES_PER_BLOCK;                // 8192
  gemm_wmma_kernel<<<blocks, 128, 0, stream>>>(x, Wp, bias, out_fcn);

  gather_kernel<<<(BATCH * 4 + 255) / 256, 256, 0, stream>>>(out_fcn, inds, out_off);
}